// GraphRes_88407606821115
// MI455X (gfx1250) — compile-verified
//
#include <hip/hip_runtime.h>
#include <math.h>

typedef __attribute__((ext_vector_type(2))) float v2f;
typedef __attribute__((ext_vector_type(8))) float v8f;

#define C1c 16
#define C2c 64
#define NV5c 380
#define NSEGc 6080   // 16 * 380
#define NX7c 16384   // 256 * 64

// ---------- helpers ----------
// Branchless ELU using the hardware transcendental (v_exp_f32).
__device__ __forceinline__ float eluf(float v) {
    float e = __expf(fminf(v, 0.f)) - 1.0f;   // unconditional: no exec divergence
    return v > 0.f ? v : e;                   // v_cndmask
}

__device__ __forceinline__ unsigned flipf(float f) {
    unsigned u = __float_as_uint(f);
    return (u & 0x80000000u) ? ~u : (u | 0x80000000u);
}
__device__ __forceinline__ float unflipf(unsigned u) {
    return __uint_as_float((u & 0x80000000u) ? (u ^ 0x80000000u) : ~u);
}
__device__ __forceinline__ void atomicMaxF(float* addr, float v) {
    unsigned u = __float_as_uint(v);
    if (u >> 31) atomicMin((unsigned*)addr, u);
    else         atomicMax((int*)addr, (int)u);
}
__device__ __forceinline__ void atomicMinF(float* addr, float v) {
    unsigned u = __float_as_uint(v);
    if (u >> 31) atomicMax((unsigned*)addr, u);
    else         atomicMin((int*)addr, (int)u);
}

// ---------- init ----------
__global__ void k_init(float* ws, int offSeg, int offPX, int offPSUM, int offX7,
                       int offStats, int total) {
    int i = blockIdx.x * blockDim.x + threadIdx.x;
    if (i >= total) return;
    float v;
    if (i < offSeg)        v = 0.f;         // s + t
    else if (i < offPX)    return;          // seg (fully written later)
    else if (i < offPSUM)  v = -INFINITY;   // px
    else if (i < offX7)    v = 0.f;         // psum + cnt
    else if (i < offStats) v = -INFINITY;   // x7
    else {
        int k = i - offStats;
        if (k < 32)       v = 0.f;          // sum1, sumsq1
        else if (k < 48)  v = -INFINITY;    // max1
        else if (k < 64)  v = INFINITY;     // min1
        else if (k < 352) v = 0.f;          // scale/shift1, sum2/sumsq2, scale/shift2
        else              v = -INFINITY;    // gmax2
    }
    ws[i] = v;
}

// ---------- stage 1 edge pass: s[dst] += x[src] ----------
__global__ void k_edge1(const float* __restrict__ x, const int* __restrict__ ei,
                        int E, float* __restrict__ s) {
    int e = blockIdx.x * blockDim.x + threadIdx.x;
    if (e >= E) return;
    atomicAdd(&s[ei[E + e]], x[ei[e]]);
}

// ---------- stage 1 node pass: bn1 stats + voxel seg + pos sums ----------
__global__ void k_node1(const float* __restrict__ s, const float* __restrict__ pos,
                        const int* __restrict__ batch, const float* __restrict__ W1,
                        const float* __restrict__ b1, int N,
                        int* __restrict__ seg, float* __restrict__ psum,
                        float* __restrict__ stats) {
    __shared__ float    bsum[C1c], bsq[C1c];
    __shared__ unsigned bmax[C1c], bmin[C1c];
    int tid = threadIdx.x;
    if (tid < C1c) {
        bsum[tid] = 0.f; bsq[tid] = 0.f;
        bmax[tid] = flipf(-INFINITY); bmin[tid] = flipf(INFINITY);
    }
    __syncthreads();
    int i = blockIdx.x * blockDim.x + tid;
    if (i < N) {
        float sv = s[i];
#pragma unroll
        for (int c = 0; c < C1c; c++) {
            float h = eluf(fmaf(sv, W1[c], b1[c]));
            atomicAdd(&bsum[c], h);
            atomicAdd(&bsq[c], h * h);
            unsigned fk = flipf(h);
            atomicMax(&bmax[c], fk);
            atomicMin(&bmin[c], fk);
        }
        float px_ = pos[3 * i], py_ = pos[3 * i + 1], pz_ = pos[3 * i + 2];
        int vx = (int)floorf(px_ / 16.0f);
        int vy = (int)floorf(py_ / 12.0f);
        int sg = batch[i] * NV5c + vx * 20 + vy;
        seg[i] = sg;
        atomicAdd(&psum[sg * 4 + 0], px_);
        atomicAdd(&psum[sg * 4 + 1], py_);
        atomicAdd(&psum[sg * 4 + 2], pz_);
        atomicAdd(&psum[sg * 4 + 3], 1.0f);
    }
    __syncthreads();
    if (tid < C1c) {
        atomicAdd(&stats[tid], bsum[tid]);
        atomicAdd(&stats[16 + tid], bsq[tid]);
        atomicMaxF(&stats[32 + tid], unflipf(bmax[tid]));
        atomicMinF(&stats[48 + tid], unflipf(bmin[tid]));
    }
}

// ---------- finalize bn1 + out1 head ----------
__global__ void k_finalize1(const float* __restrict__ bn1_g, const float* __restrict__ bn1_b,
                            const float* __restrict__ ln1_g, const float* __restrict__ ln1_b,
                            const float* __restrict__ fc1_W, int N,
                            float* __restrict__ stats, float* __restrict__ out) {
    __shared__ float xm[C1c];
    int tid = threadIdx.x;
    if (tid < C1c) {
        float m   = stats[tid] / (float)N;
        float var = stats[16 + tid] / (float)N - m * m;
        float sc  = rsqrtf(var + 1e-5f) * bn1_g[tid];
        float sh  = bn1_b[tid] - m * sc;
        stats[64 + tid] = sc;
        stats[80 + tid] = sh;
        float hx = (sc >= 0.f) ? stats[32 + tid] : stats[48 + tid];
        xm[tid] = hx * sc + sh;
    }
    __syncthreads();
    if (tid == 0) {
        float m = 0.f;
        for (int c = 0; c < C1c; c++) m += xm[c];
        m *= (1.0f / 16.0f);
        float v = 0.f;
        for (int c = 0; c < C1c; c++) { float d = xm[c] - m; v += d * d; }
        v *= (1.0f / 16.0f);
        float rs = rsqrtf(v + 1e-5f);
        float y[C1c];
        for (int c = 0; c < C1c; c++) y[c] = (xm[c] - m) * rs * ln1_g[c] + ln1_b[c];
        for (int o = 0; o < 2; o++) {
            float acc = 0.f;
            for (int c = 0; c < C1c; c++) acc += y[c] * fc1_W[o * 16 + c];
            out[o] = acc;
        }
    }
}

// ---------- stage 2 edge pass: t[dst] += x1(src) (x1 recomputed from scalar s) ----------
__global__ void k_edge2(const float* __restrict__ s, const int* __restrict__ ei, int E,
                        const float* __restrict__ W1, const float* __restrict__ b1,
                        const float* __restrict__ stats, float* __restrict__ t) {
    int e = blockIdx.x * blockDim.x + threadIdx.x;
    if (e >= E) return;
    int src = ei[e], dst = ei[E + e];
    float sv = s[src];
    float* trow = t + (size_t)dst * 16;
#pragma unroll
    for (int c = 0; c < C1c; c++) {
        float h  = eluf(fmaf(sv, W1[c], b1[c]));
        float x1 = fmaf(h, stats[64 + c], stats[80 + c]);
        atomicAdd(&trow[c], x1);
    }
}

// ---------- WMMA GEMM t[N,16] @ W2[16,64], MODE 0 = bn2 stats, MODE 1 = bn+pool5 scatter ----------
template <int MODE>
__global__ void k_gemm(const float* __restrict__ t, const float* __restrict__ W2,
                       const float* __restrict__ b2, const float* __restrict__ scale2,
                       const float* __restrict__ shift2, const int* __restrict__ seg,
                       int N, float* __restrict__ sum2, float* __restrict__ sumsq2,
                       float* __restrict__ px) {
    __shared__ float blkS[C2c], blkQ[C2c];
    int tid = threadIdx.x;
    if (MODE == 0) {
        if (tid < C2c) { blkS[tid] = 0.f; blkQ[tid] = 0.f; }
        __syncthreads();
    }
    int lane   = tid & 31;
    int gwid   = (blockIdx.x * blockDim.x + tid) >> 5;
    int nwaves = (gridDim.x * blockDim.x) >> 5;
    int half   = lane >> 4;   // 0: lanes 0-15, 1: lanes 16-31
    int nn     = lane & 15;

    // B fragments (constant across tiles): lane = N column, VGPRs/halves = K
    v2f bf[4][4];
#pragma unroll
    for (int kt = 0; kt < 4; kt++) {
        int k0 = kt * 4 + 2 * half;
#pragma unroll
        for (int nt = 0; nt < 4; nt++) {
            int ch = nt * 16 + nn;
            bf[kt][nt].x = W2[k0 * 64 + ch];
            bf[kt][nt].y = W2[(k0 + 1) * 64 + ch];
        }
    }
    float bias[4], sc[4], sh[4];
#pragma unroll
    for (int nt = 0; nt < 4; nt++) {
        int ch = nt * 16 + nn;
        bias[nt] = b2[ch];
        if (MODE == 1) { sc[nt] = scale2[ch]; sh[nt] = shift2[ch]; }
    }

    float psv[4] = {0, 0, 0, 0}, pqv[4] = {0, 0, 0, 0};
    int ntiles = N >> 4;
    for (int tile = gwid; tile < ntiles; tile += nwaves) {
        int node = tile * 16 + nn;
        const float* trow = t + (size_t)node * 16 + 2 * half;
        v2f af[4];
#pragma unroll
        for (int kt = 0; kt < 4; kt++) af[kt] = *(const v2f*)(trow + kt * 4);

        int sgv[8];
        if (MODE == 1) {
#pragma unroll
            for (int v = 0; v < 8; v++) sgv[v] = seg[tile * 16 + v + 8 * half];
        }
#pragma unroll
        for (int nt = 0; nt < 4; nt++) {
            v8f c = {0.f, 0.f, 0.f, 0.f, 0.f, 0.f, 0.f, 0.f};
#pragma unroll
            for (int kt = 0; kt < 4; kt++) {
                c = __builtin_amdgcn_wmma_f32_16x16x4_f32(
                        false, af[kt], false, bf[kt][nt], (short)0, c, false, false);
            }
#pragma unroll
            for (int v = 0; v < 8; v++) {
                float val = eluf(c[v] + bias[nt]);
                if (MODE == 0) {
                    psv[nt] += val;
                    pqv[nt] += val * val;
                } else {
                    float x2 = fmaf(val, sc[nt], sh[nt]);
                    atomicMaxF(&px[(size_t)sgv[v] * 64 + nt * 16 + nn], x2);
                }
            }
        }
    }
    if (MODE == 0) {
#pragma unroll
        for (int nt = 0; nt < 4; nt++) {
            atomicAdd(&blkS[nt * 16 + nn], psv[nt]);
            atomicAdd(&blkQ[nt * 16 + nn], pqv[nt]);
        }
        __syncthreads();
        if (tid < C2c) {
            atomicAdd(&sum2[tid], blkS[tid]);
            atomicAdd(&sumsq2[tid], blkQ[tid]);
        }
    }
}

// ---------- finalize bn2 params ----------
__global__ void k_finalize2(const float* __restrict__ bn2_g, const float* __restrict__ bn2_b,
                            int N, float* __restrict__ stats) {
    int c = threadIdx.x;
    if (c < C2c) {
        float m   = stats[96 + c] / (float)N;
        float var = stats[160 + c] / (float)N - m * m;
        float sc  = rsqrtf(var + 1e-5f) * bn2_g[c];
        stats[224 + c] = sc;
        stats[288 + c] = bn2_b[c] - m * sc;
    }
}

// ---------- pool7 scatter + global channel max for out2 ----------
__global__ void k_pool7(const float* __restrict__ px, const float* __restrict__ psum,
                        float* __restrict__ x7, float* __restrict__ gmax2, int nseg) {
    int sg = blockIdx.x * blockDim.x + threadIdx.x;
    if (sg >= nseg) return;
    float cnt = psum[sg * 4 + 3];
    float ppx = 0.f, ppy = 0.f;
    if (cnt > 0.f) {
        float inv = 1.0f / fmaxf(cnt, 1.0f);
        ppx = psum[sg * 4 + 0] * inv;
        ppy = psum[sg * 4 + 1] * inv;
    }
    int v7x  = min(max((int)floorf(ppx / 101.0f), 0), 3);
    int v7y  = min(max((int)floorf(ppy / 80.0f), 0), 2);
    int idx7 = (sg / NV5c) * 16 + (v7x * 3 + v7y);
    const float* row = px + (size_t)sg * 64;
    float* x7row = x7 + (size_t)idx7 * 64;
    for (int c = 0; c < C2c; c++) {
        float v = row[c];
        atomicMaxF(&x7row[c], v);
        atomicMaxF(&gmax2[c], v);
    }
}

// ---------- final heads: out2 and out ----------
__global__ void k_final(const float* __restrict__ x7, const float* __restrict__ gmax2,
                        const float* __restrict__ ln2_g, const float* __restrict__ ln2_b,
                        const float* __restrict__ fc2_W, const float* __restrict__ fcW,
                        float* __restrict__ out) {
    int tid = threadIdx.x;
    if (tid < 32) {
        int b = tid >> 1, o = tid & 1;
        float acc = 0.f;
        for (int j = 0; j < 1024; j++) {
            float v = x7[b * 1024 + j];
            if (v == -INFINITY) v = 0.f;   // torch_scatter zero-fill
            acc += v * fcW[o * 1024 + j];
        }
        out[4 + b * 2 + o] = acc;
    } else if (tid == 32 || tid == 33) {
        int o = tid - 32;
        float m = 0.f;
        for (int c = 0; c < C2c; c++) m += gmax2[c];
        m *= (1.0f / 64.0f);
        float v = 0.f;
        for (int c = 0; c < C2c; c++) { float d = gmax2[c] - m; v += d * d; }
        v *= (1.0f / 64.0f);
        float rs = rsqrtf(v + 1e-5f);
        float acc = 0.f;
        for (int c = 0; c < C2c; c++) {
            float y = (gmax2[c] - m) * rs * ln2_g[c] + ln2_b[c];
            acc += y * fc2_W[o * 64 + c];
        }
        out[2 + o] = acc;
    }
}

extern "C" void kernel_launch(void* const* d_in, const int* in_sizes, int n_in,
                              void* d_out, int out_size, void* d_ws, size_t ws_size,
                              hipStream_t stream) {
    const float* x     = (const float*)d_in[0];
    const float* pos   = (const float*)d_in[1];
    const float* W1    = (const float*)d_in[2];
    const float* b1    = (const float*)d_in[3];
    const float* W2    = (const float*)d_in[4];
    const float* b2    = (const float*)d_in[5];
    const float* bn1_g = (const float*)d_in[6];
    const float* bn1_b = (const float*)d_in[7];
    const float* bn2_g = (const float*)d_in[8];
    const float* bn2_b = (const float*)d_in[9];
    const float* ln1_g = (const float*)d_in[10];
    const float* ln1_b = (const float*)d_in[11];
    const float* ln2_g = (const float*)d_in[12];
    const float* ln2_b = (const float*)d_in[13];
    const float* fc1_W = (const float*)d_in[14];
    const float* fc2_W = (const float*)d_in[15];
    const float* fc_W  = (const float*)d_in[16];
    const int*   ei    = (const int*)d_in[17];
    const int*   batch = (const int*)d_in[18];

    const int N = in_sizes[0];
    const int E = in_sizes[17] / 2;

    float* ws = (float*)d_ws;
    const int offS     = 0;
    const int offT     = offS + N;
    const int offSeg   = offT + N * 16;
    const int offPX    = offSeg + N;
    const int offPSUM  = offPX + NSEGc * 64;
    const int offX7    = offPSUM + NSEGc * 4;
    const int offStats = offX7 + NX7c;
    const int total    = offStats + 416;

    float* s     = ws + offS;
    float* t     = ws + offT;
    int*   seg   = (int*)(ws + offSeg);
    float* px    = ws + offPX;
    float* psum  = ws + offPSUM;
    float* x7    = ws + offX7;
    float* stats = ws + offStats;
    float* out   = (float*)d_out;

    const int TB = 256;
    k_init<<<(total + TB - 1) / TB, TB, 0, stream>>>(ws, offSeg, offPX, offPSUM,
                                                     offX7, offStats, total);
    k_edge1<<<(E + TB - 1) / TB, TB, 0, stream>>>(x, ei, E, s);
    k_node1<<<(N + TB - 1) / TB, TB, 0, stream>>>(s, pos, batch, W1, b1, N, seg, psum, stats);
    k_finalize1<<<1, 32, 0, stream>>>(bn1_g, bn1_b, ln1_g, ln1_b, fc1_W, N, stats, out);
    k_edge2<<<(E + TB - 1) / TB, TB, 0, stream>>>(s, ei, E, W1, b1, stats, t);
    // WMMA pass 1: bn2 statistics
    k_gemm<0><<<512, TB, 0, stream>>>(t, W2, b2, stats + 224, stats + 288, seg, N,
                                      stats + 96, stats + 160, px);
    k_finalize2<<<1, 64, 0, stream>>>(bn2_g, bn2_b, N, stats);
    // WMMA pass 2: bn affine + pool5 max-scatter
    k_gemm<1><<<512, TB, 0, stream>>>(t, W2, b2, stats + 224, stats + 288, seg, N,
                                      stats + 96, stats + 160, px);
    k_pool7<<<(NSEGc + TB - 1) / TB, TB, 0, stream>>>(px, psum, x7, stats + 352, NSEGc);
    k_final<<<1, 64, 0, stream>>>(x7, stats + 352, ln2_g, ln2_b, fc2_W, fc_W, out);
}